// mLSTM_14903536517285
// MI455X (gfx1250) — compile-verified
//
#include <hip/hip_runtime.h>

typedef float v2f __attribute__((ext_vector_type(2)));
typedef float v4f __attribute__((ext_vector_type(4)));
typedef float v8f __attribute__((ext_vector_type(8)));

#define D_IN   128       // input feature dim
#define MDIM   64        // mem_dim == hidden
#define TT     16        // tokens per block (one WMMA M-tile)
#define NTHR   128       // 4 waves of 32
#define T_TOT  (8 * 2048)

// One 16x16 f32 output tile of X[16x128] @ W[128x64], columns [colbase, colbase+16)
// K accumulated 4 at a time with V_WMMA_F32_16X16X4_F32.
__device__ __forceinline__ v8f proj16(const float* xl, const float* __restrict__ W,
                                      int idx16, int hi, int colbase) {
  v8f acc = {};
  #pragma unroll
  for (int k0 = 0; k0 < D_IN; k0 += 4) {
    const int kk = k0 + 2 * hi;
    v2f a;                                      // A frag: row = idx16, K = kk, kk+1
    a.x = xl[idx16 * D_IN + kk];
    a.y = xl[idx16 * D_IN + kk + 1];
    v2f b;                                      // B frag: col = colbase+idx16
    b.x = W[kk * MDIM + colbase + idx16];
    b.y = W[(kk + 1) * MDIM + colbase + idx16];
    acc = __builtin_amdgcn_wmma_f32_16x16x4_f32(false, a, false, b, (short)0, acc,
                                                false, false);
  }
  return acc;
}

__global__ __launch_bounds__(NTHR) void mlstm_fused(
    const float* __restrict__ x, const float* __restrict__ c_prev,
    const float* __restrict__ n_prev,
    const float* __restrict__ wq, const float* __restrict__ bq,
    const float* __restrict__ wk, const float* __restrict__ bk,
    const float* __restrict__ wv, const float* __restrict__ bv,
    const float* __restrict__ wi, const float* __restrict__ bi,
    const float* __restrict__ wf, const float* __restrict__ bf,
    const float* __restrict__ wo, const float* __restrict__ bo,
    float* __restrict__ h_out, float* __restrict__ c_out,
    float* __restrict__ n_out)
{
  __shared__ float x_lds[TT * D_IN];      // 8 KB   x tile
  __shared__ float cp_lds[MDIM * MDIM];   // 16 KB  c_prev
  __shared__ float q_lds[TT * MDIM];      // 4 KB each
  __shared__ float k_lds[TT * MDIM];      // (k already scaled by 1/sqrt(M))
  __shared__ float v_lds[TT * MDIM];
  __shared__ float o_lds[TT * MDIM];      // pre-activation
  __shared__ float cpq_lds[TT * MDIM];    // (c_prev @ q) per token
  __shared__ float np_lds[MDIM];
  __shared__ float it_lds[TT], ft_lds[TT], kq_lds[TT], nq_lds[TT];

  const int tid    = threadIdx.x;
  const int lane   = tid & 31;
  const int wave   = tid >> 5;
  const int idx16  = lane & 15;
  const int hi     = lane >> 4;
  const int colbase = wave * 16;          // 4 waves cover 64 columns
  const int tok0   = blockIdx.x * TT;

  // ---- Stage x tile, c_prev, n_prev into LDS ----
  {
    const v4f* xg = (const v4f*)(x + (size_t)tok0 * D_IN);
    v4f* xl = (v4f*)x_lds;
    for (int i = tid; i < TT * D_IN / 4; i += NTHR) xl[i] = xg[i];
    const v4f* cg = (const v4f*)c_prev;
    v4f* cl = (v4f*)cp_lds;
    for (int i = tid; i < MDIM * MDIM / 4; i += NTHR) cl[i] = cg[i];
    if (tid < MDIM) np_lds[tid] = n_prev[tid];
  }
  __syncthreads();

  // ---- Projections via f32 WMMA (each wave: one 16-col tile per matrix) ----
  {
    v8f acc = proj16(x_lds, wq, idx16, hi, colbase);
    const float bias = bq[colbase + idx16];
    #pragma unroll
    for (int r = 0; r < 8; ++r)
      q_lds[(r + 8 * hi) * MDIM + colbase + idx16] = acc[r] + bias;
  }
  {
    v8f acc = proj16(x_lds, wk, idx16, hi, colbase);
    const float bias = bk[colbase + idx16];
    #pragma unroll
    for (int r = 0; r < 8; ++r)                       // scale by 1/sqrt(64)
      k_lds[(r + 8 * hi) * MDIM + colbase + idx16] = (acc[r] + bias) * 0.125f;
  }
  {
    v8f acc = proj16(x_lds, wv, idx16, hi, colbase);
    const float bias = bv[colbase + idx16];
    #pragma unroll
    for (int r = 0; r < 8; ++r)
      v_lds[(r + 8 * hi) * MDIM + colbase + idx16] = acc[r] + bias;
  }
  {
    v8f acc = proj16(x_lds, wo, idx16, hi, colbase);
    const float bias = bo[colbase + idx16];
    #pragma unroll
    for (int r = 0; r < 8; ++r)
      o_lds[(r + 8 * hi) * MDIM + colbase + idx16] = acc[r] + bias;
  }
  __syncthreads();

  // ---- Scalar gates + per-token dot products (16 lanes, others idle) ----
  if (tid < TT) {
    const int t = tid;
    float ai = 0.f, af = 0.f;
    for (int d = 0; d < D_IN; ++d) {
      const float xv = x_lds[t * D_IN + d];
      ai += xv * wi[d];
      af += xv * wf[d];
    }
    it_lds[t] = __expf(ai + bi[0]);
    ft_lds[t] = 1.0f / (1.0f + __expf(-(af + bf[0])));
    float kq = 0.f, nq = 0.f;
    for (int m = 0; m < MDIM; ++m) {
      const float qv = q_lds[t * MDIM + m];
      kq += k_lds[t * MDIM + m] * qv;   // k^T q  (k pre-scaled)
      nq += np_lds[m] * qv;             // n_prev^T q
    }
    kq_lds[t] = kq;
    nq_lds[t] = nq;
  }

  // ---- cpq = Q[16x64] @ c_prev^T  (result[t][m] = sum_k q[t][k] c_prev[m][k]) ----
  {
    v8f acc = {};
    #pragma unroll
    for (int k0 = 0; k0 < MDIM; k0 += 4) {
      const int kk = k0 + 2 * hi;
      v2f a;
      a.x = q_lds[idx16 * MDIM + kk];
      a.y = q_lds[idx16 * MDIM + kk + 1];
      v2f b;                                       // B[k][m] = c_prev[m][k]
      b.x = cp_lds[(colbase + idx16) * MDIM + kk];
      b.y = cp_lds[(colbase + idx16) * MDIM + kk + 1];
      acc = __builtin_amdgcn_wmma_f32_16x16x4_f32(false, a, false, b, (short)0, acc,
                                                  false, false);
    }
    #pragma unroll
    for (int r = 0; r < 8; ++r)
      cpq_lds[(r + 8 * hi) * MDIM + colbase + idx16] = acc[r];
  }
  __syncthreads();

  // ---- h_t and n_t (small) ----
  for (int i = tid; i < TT * MDIM; i += NTHR) {
    const int t = i >> 6, m = i & 63;
    const float it = it_lds[t], ft = ft_lds[t];
    const float kq = kq_lds[t];
    n_out[(size_t)(tok0 + t) * MDIM + m] = ft * np_lds[m] + it * k_lds[t * MDIM + m];
    const float denom = fmaxf(fabsf(ft * nq_lds[t] + it * kq), 1.0f);
    const float htil  = (ft * cpq_lds[t * MDIM + m] + it * v_lds[t * MDIM + m] * kq) / denom;
    const float o     = 1.0f / (1.0f + __expf(-o_lds[t * MDIM + m]));
    h_out[(size_t)(tok0 + t) * MDIM + m] = o * htil;
  }

  // ---- c_t: the 256 MB stream. Non-temporal 128-bit stores. ----
  for (int t = 0; t < TT; ++t) {
    const float it = it_lds[t], ft = ft_lds[t];
    const v4f* cpv = (const v4f*)cp_lds;
    const v4f* krow = (const v4f*)&k_lds[t * MDIM];
    v4f* cdst = (v4f*)(c_out + (size_t)(tok0 + t) * MDIM * MDIM);
    for (int i = tid; i < MDIM * MDIM / 4; i += NTHR) {
      const int m = i >> 4;                    // row of the 64x64 tile
      const float ivm = it * v_lds[t * MDIM + m];
      const v4f kk = krow[i & 15];             // 4 consecutive k values
      v4f r = cpv[i] * ft + kk * ivm;
      __builtin_nontemporal_store(r, &cdst[i]);
    }
  }
}

extern "C" void kernel_launch(void* const* d_in, const int* in_sizes, int n_in,
                              void* d_out, int out_size, void* d_ws, size_t ws_size,
                              hipStream_t stream) {
  const float* x      = (const float*)d_in[0];
  const float* c_prev = (const float*)d_in[1];
  const float* n_prev = (const float*)d_in[2];
  const float* wq = (const float*)d_in[3];
  const float* bq = (const float*)d_in[4];
  const float* wk = (const float*)d_in[5];
  const float* bk = (const float*)d_in[6];
  const float* wv = (const float*)d_in[7];
  const float* bv = (const float*)d_in[8];
  const float* wi = (const float*)d_in[9];
  const float* bi = (const float*)d_in[10];
  const float* wf = (const float*)d_in[11];
  const float* bf = (const float*)d_in[12];
  const float* wo = (const float*)d_in[13];
  const float* bo = (const float*)d_in[14];

  float* h_out = (float*)d_out;                                  // [T, 64]
  float* c_out = h_out + (size_t)T_TOT * MDIM;                   // [T, 64, 64]
  float* n_out = c_out + (size_t)T_TOT * MDIM * MDIM;            // [T, 64]

  dim3 grid(T_TOT / TT), block(NTHR);
  hipLaunchKernelGGL(mlstm_fused, grid, block, 0, stream,
                     x, c_prev, n_prev, wq, bq, wk, bk, wv, bv,
                     wi, bi, wf, bf, wo, bo, h_out, c_out, n_out);
}